// DigitCaps_2594160247456
// MI455X (gfx1250) — compile-verified
//
#include <hip/hip_runtime.h>

typedef float v2f __attribute__((ext_vector_type(2)));
typedef float v8f __attribute__((ext_vector_type(8)));

#define BATCH   128
#define NROUTES 4608
#define NCAPS   16
#define OUT_CH  16
#define IN_CH   8
#define RPB     32                 // routes per wave in WMMA kernels
#define RBLKS   (NROUTES / RPB)    // 144
#define BTILES  (BATCH / 16)       // 8
#define S_ELEMS (BATCH * NCAPS * OUT_CH)   // 32768
#define BL_ELEMS (NCAPS * NROUTES)         // 73728

#define WMMA_F32(A, B, C) \
    __builtin_amdgcn_wmma_f32_16x16x4_f32(false, (A), false, (B), (short)0, (C), false, false)

// ---------------- x transpose: [B,R,I] -> [R,I,B] (coalesced WMMA B loads) ---
__global__ void k_transpose_x(const float* __restrict__ x, float* __restrict__ xT) {
    int idx = blockIdx.x * blockDim.x + threadIdx.x;
    if (idx >= NROUTES * IN_CH * BATCH) return;
    int b = idx & (BATCH - 1);
    int i = (idx >> 7) & (IN_CH - 1);
    int r = idx >> 10;
    xT[idx] = x[(b * NROUTES + r) * IN_CH + i];
}

__global__ void k_zero(float* __restrict__ p, int n) {
    int idx = blockIdx.x * blockDim.x + threadIdx.x;
    if (idx < n) p[idx] = 0.0f;
}

// ---------------- softmax over routes, per capsule; bl/cc layout [C][R] ------
__global__ void k_softmax(const float* __restrict__ bl, float* __restrict__ cc) {
    __shared__ float red[256];
    const int cap = blockIdx.x;
    const int tid = threadIdx.x;
    const float* col = bl + cap * NROUTES;
    float vals[NROUTES / 256];
    float m = -3.0e38f;
    #pragma unroll
    for (int k = 0; k < NROUTES / 256; ++k) {
        vals[k] = col[tid + k * 256];
        m = fmaxf(m, vals[k]);
    }
    red[tid] = m; __syncthreads();
    for (int s = 128; s > 0; s >>= 1) {
        if (tid < s) red[tid] = fmaxf(red[tid], red[tid + s]);
        __syncthreads();
    }
    m = red[0]; __syncthreads();
    float sum = 0.0f;
    #pragma unroll
    for (int k = 0; k < NROUTES / 256; ++k) {
        vals[k] = __expf(vals[k] - m);
        sum += vals[k];
    }
    red[tid] = sum; __syncthreads();
    for (int s = 128; s > 0; s >>= 1) {
        if (tid < s) red[tid] += red[tid + s];
        __syncthreads();
    }
    const float inv = 1.0f / red[0];
    float* out = cc + cap * NROUTES;
    #pragma unroll
    for (int k = 0; k < NROUTES / 256; ++k) out[tid + k * 256] = vals[k] * inv;
}

// ---------------- fused u_hat + weighted route-sum, split-K over route blocks.
// One wave per (route-block, capsule-pair, batch-tile-pair): 2x2 register
// blocking -> 8 WMMA per route at 12 VMEM (x shared across caps, W across bt).
__global__ void k_uhat_s(const float* __restrict__ W, const float* __restrict__ xT,
                         const float* __restrict__ cc, float* __restrict__ s_part) {
    const int lane = threadIdx.x;          // full EXEC (WMMA requirement)
    const int rb   = blockIdx.x;
    const int cap0 = blockIdx.y * 2;
    const int b0   = blockIdx.z * 32;      // two 16-batch tiles
    const int n    = lane & 15;            // A: M index (o); B: N index (batch)
    const int h    = lane >> 4;
    const int k0   = 2 * h;                // per-lane K pair (ISA 16x16x4 layout)

    v8f acc[2][2];
    #pragma unroll
    for (int ci = 0; ci < 2; ++ci)
        #pragma unroll
        for (int bi = 0; bi < 2; ++bi)
            acc[ci][bi] = (v8f){0.f, 0.f, 0.f, 0.f, 0.f, 0.f, 0.f, 0.f};

    const float* cc0 = cc + (cap0 + 0) * NROUTES;
    const float* cc1 = cc + (cap0 + 1) * NROUTES;
    const int r0 = rb * RPB;
    #pragma unroll 2
    for (int rr = 0; rr < RPB; ++rr) {
        const int r = r0 + rr;
        const float cr0 = cc0[r];          // wave-uniform -> scalar loads
        const float cr1 = cc1[r];
        const float* wrow = W + ((size_t)(r * NCAPS + cap0) * OUT_CH + n) * IN_CH;
        // A operands (scaled W) for both capsules
        v2f a0_1 = { wrow[k0] * cr0,     wrow[k0 + 1] * cr0 };
        v2f a0_2 = { wrow[4 + k0] * cr0, wrow[5 + k0] * cr0 };
        const float* wrow1 = wrow + OUT_CH * IN_CH;
        v2f a1_1 = { wrow1[k0] * cr1,     wrow1[k0 + 1] * cr1 };
        v2f a1_2 = { wrow1[4 + k0] * cr1, wrow1[5 + k0] * cr1 };
        // B operands (xT) for both batch tiles, shared across capsules
        const float* xc = xT + (size_t)r * (IN_CH * BATCH) + b0 + n;
        v2f b0_1 = { xc[k0 * BATCH],            xc[(k0 + 1) * BATCH] };
        v2f b0_2 = { xc[(4 + k0) * BATCH],      xc[(5 + k0) * BATCH] };
        v2f b1_1 = { xc[k0 * BATCH + 16],       xc[(k0 + 1) * BATCH + 16] };
        v2f b1_2 = { xc[(4 + k0) * BATCH + 16], xc[(5 + k0) * BATCH + 16] };

        acc[0][0] = WMMA_F32(a0_1, b0_1, acc[0][0]);
        acc[0][0] = WMMA_F32(a0_2, b0_2, acc[0][0]);
        acc[0][1] = WMMA_F32(a0_1, b1_1, acc[0][1]);
        acc[0][1] = WMMA_F32(a0_2, b1_2, acc[0][1]);
        acc[1][0] = WMMA_F32(a1_1, b0_1, acc[1][0]);
        acc[1][0] = WMMA_F32(a1_2, b0_2, acc[1][0]);
        acc[1][1] = WMMA_F32(a1_1, b1_1, acc[1][1]);
        acc[1][1] = WMMA_F32(a1_2, b1_2, acc[1][1]);
    }
    // D layout: VGPR j -> (m = j + 8h, n); write disjoint split-K partials
    #pragma unroll
    for (int ci = 0; ci < 2; ++ci) {
        #pragma unroll
        for (int bi = 0; bi < 2; ++bi) {
            float* dst = s_part + (size_t)rb * S_ELEMS
                       + (b0 + bi * 16 + n) * (NCAPS * OUT_CH)
                       + (cap0 + ci) * OUT_CH + 8 * h;
            #pragma unroll
            for (int j = 0; j < 8; ++j) dst[j] = acc[ci][bi][j];
        }
    }
}

__global__ void k_reduce_s(const float* __restrict__ s_part, float* __restrict__ s) {
    int idx = blockIdx.x * blockDim.x + threadIdx.x;
    if (idx >= S_ELEMS) return;
    float acc = 0.0f;
    #pragma unroll 4
    for (int rb = 0; rb < RBLKS; ++rb) acc += s_part[(size_t)rb * S_ELEMS + idx];
    s[idx] = acc;
}

// ---------------- squash along O --------------------------------------------
__global__ void k_squash(const float* __restrict__ s, float* __restrict__ v) {
    int idx = blockIdx.x * blockDim.x + threadIdx.x;   // (b, cap)
    if (idx >= BATCH * NCAPS) return;
    const float* sp = s + idx * OUT_CH;
    float t[OUT_CH];
    float sn = 0.0f;
    #pragma unroll
    for (int j = 0; j < OUT_CH; ++j) { t[j] = sp[j]; sn += t[j] * t[j]; }
    const float scale = sn / ((1.0f + sn) * sqrtf(sn));
    float* vp = v + idx * OUT_CH;
    #pragma unroll
    for (int j = 0; j < OUT_CH; ++j) vp[j] = t[j] * scale;
}

// ---------------- agreement: recompute u_hat tile, dot with v ---------------
// Per route: 2 WMMA + 8 FMA + one xor-16 shuffle; lanes 0-15 then hold the
// full per-batch dot (o-halves folded) and store one coalesced 64B row.
__global__ void k_agree(const float* __restrict__ W, const float* __restrict__ xT,
                        const float* __restrict__ v, float* __restrict__ a_part) {
    const int lane = threadIdx.x;
    const int rb   = blockIdx.x;
    const int cap  = blockIdx.y;
    const int bt   = blockIdx.z;
    const int b0   = bt * 16;
    const int n    = lane & 15;
    const int h    = lane >> 4;
    const int k0   = 2 * h;

    // v values matching this lane's D elements: (b0+n, cap, 8h + j)
    float vv[8];
    const float* vp = v + (b0 + n) * (NCAPS * OUT_CH) + cap * OUT_CH + 8 * h;
    #pragma unroll
    for (int j = 0; j < 8; ++j) vv[j] = vp[j];

    float* adst = a_part + ((size_t)(bt * NCAPS + cap) * NROUTES) * 16;
    const int r0 = rb * RPB;
    #pragma unroll 2
    for (int rr = 0; rr < RPB; ++rr) {
        const int r = r0 + rr;
        const float* wrow = W + ((size_t)(r * NCAPS + cap) * OUT_CH + n) * IN_CH;
        v2f a1 = { wrow[k0],     wrow[k0 + 1] };
        v2f a2 = { wrow[4 + k0], wrow[5 + k0] };
        const float* xc = xT + (size_t)r * (IN_CH * BATCH) + b0 + n;
        v2f b1 = { xc[k0 * BATCH],       xc[(k0 + 1) * BATCH] };
        v2f b2 = { xc[(4 + k0) * BATCH], xc[(5 + k0) * BATCH] };
        v8f acc = {0.f, 0.f, 0.f, 0.f, 0.f, 0.f, 0.f, 0.f};
        acc = WMMA_F32(a1, b1, acc);
        acc = WMMA_F32(a2, b2, acc);
        float dot = 0.0f;
        #pragma unroll
        for (int j = 0; j < 8; ++j) dot += acc[j] * vv[j];
        dot += __shfl_xor(dot, 16, 32);    // fold o=8..15 half onto lanes 0..15
        if (lane < 16) adst[(size_t)r * 16 + n] = dot;
    }
}

// bl[cap][r] += mean over batch of agreement partials (8 bt x 16 bb each)
__global__ void k_update_bl(const float* __restrict__ a_part, float* __restrict__ bl) {
    int idx = blockIdx.x * blockDim.x + threadIdx.x;   // [cap][r]
    if (idx >= BL_ELEMS) return;
    const int cap = idx / NROUTES;
    const int r   = idx - cap * NROUTES;
    float acc = 0.0f;
    #pragma unroll
    for (int t = 0; t < BTILES; ++t) {
        const float* ap = a_part + ((size_t)(t * NCAPS + cap) * NROUTES + r) * 16;
        #pragma unroll
        for (int j = 0; j < 16; ++j) acc += ap[j];
    }
    bl[idx] += acc * (1.0f / BATCH);
}

extern "C" void kernel_launch(void* const* d_in, const int* in_sizes, int n_in,
                              void* d_out, int out_size, void* d_ws, size_t ws_size,
                              hipStream_t stream) {
    const float* x = (const float*)d_in[0];   // [B, R, I]
    const float* W = (const float*)d_in[1];   // [1, R, C, O, I]
    float* ws = (float*)d_ws;

    float* xT     = ws;                                        // 4,718,592
    float* s_part = xT + (size_t)NROUTES * IN_CH * BATCH;      // 144*32768 = 4,718,592
    float* s      = s_part + (size_t)RBLKS * S_ELEMS;          // 32,768
    float* v      = s + S_ELEMS;                               // 32,768
    float* bl     = v + S_ELEMS;                               // 73,728
    float* cc     = bl + BL_ELEMS;                             // 73,728
    float* a_part = cc + BL_ELEMS;                             // 8*16*4608*16 = 9,437,184

    k_transpose_x<<<(NROUTES * IN_CH * BATCH + 255) / 256, 256, 0, stream>>>(x, xT);
    k_zero<<<(BL_ELEMS + 255) / 256, 256, 0, stream>>>(bl, BL_ELEMS);

    dim3 gs(RBLKS, NCAPS / 2, BTILES / 2);    // 144 x 8 x 4 waves (2x2 blocked)
    dim3 ga(RBLKS, NCAPS, BTILES);            // 144 x 16 x 8 waves
    for (int it = 0; it < 3; ++it) {
        k_softmax<<<NCAPS, 256, 0, stream>>>(bl, cc);
        k_uhat_s<<<gs, 32, 0, stream>>>(W, xT, cc, s_part);
        k_reduce_s<<<(S_ELEMS + 255) / 256, 256, 0, stream>>>(s_part, s);
        float* vdst = (it == 2) ? (float*)d_out : v;
        k_squash<<<(BATCH * NCAPS + 255) / 256, 256, 0, stream>>>(s, vdst);
        if (it < 2) {
            k_agree<<<ga, 32, 0, stream>>>(W, xT, v, a_part);
            k_update_bl<<<(BL_ELEMS + 255) / 256, 256, 0, stream>>>(a_part, bl);
        }
    }
}